// RegionProposalNetwork_90263032693147
// MI455X (gfx1250) — compile-verified
//
#include <hip/hip_runtime.h>
#include <math.h>

// ---- problem constants (match reference) ----
#define A_     9
#define HW_    4096        // 64*64
#define NPB    36864       // anchors per image = 64*64*9
#define B_     8
#define KPRE   2000
#define KPOST  1000
#define IOUT   0.7f
#define BCLIP  4.135166556742356f   // log(1000/16)
#define MINSZ  0.001f
#define IMGSZ  1024.0f
#define NT_    (KPRE / 16)          // 125 chunks of 16 boxes

typedef float v2f __attribute__((ext_vector_type(2)));
typedef float v8f __attribute__((ext_vector_type(8)));
typedef int   v4i __attribute__((ext_vector_type(4)));

// gfx1250 async global->LDS path (guarded; falls back to plain loads)
#if defined(__has_builtin)
#if __has_builtin(__builtin_amdgcn_global_load_async_to_lds_b128) && \
    __has_builtin(__builtin_amdgcn_s_wait_asynccnt)
#define USE_ASYNC_LDS 1
#endif
#endif
#ifndef USE_ASYNC_LDS
#define USE_ASYNC_LDS 0
#endif

#if USE_ASYNC_LDS
typedef __attribute__((address_space(1))) v4i* gas_v4i_ptr;  // global
typedef __attribute__((address_space(3))) v4i* las_v4i_ptr;  // LDS
#endif

__device__ __forceinline__ unsigned f2key(float f) {
  unsigned u = __float_as_uint(f);
  return (u & 0x80000000u) ? ~u : (u | 0x80000000u);  // monotone: bigger float -> bigger key
}

// ---------------------------------------------------------------------------
// Kernel 1: fused transpose + box decode + clip.
// scores[b][p], p = (h*64+w)*9 + a ; boxes as float4 AoS.
// ---------------------------------------------------------------------------
__global__ void rpn_decode(const float* __restrict__ obj,
                           const float* __restrict__ del,
                           const float* __restrict__ anc,
                           float4* __restrict__ boxes,
                           float* __restrict__ scores) {
  int b = blockIdx.x;
  int p = blockIdx.y * blockDim.x + threadIdx.x;
  if (p >= NPB) return;
  int a  = p % A_;
  int hw = p / A_;

  scores[(size_t)b * NPB + p] = obj[((size_t)b * A_ + a) * HW_ + hw];

  const float* dbase = del + ((size_t)(b * A_ + a) * 4) * HW_ + hw;
  float dx = dbase[0];
  float dy = dbase[HW_];
  float dw = fminf(dbase[2 * HW_], BCLIP);
  float dh = fminf(dbase[3 * HW_], BCLIP);

  float4 an = ((const float4*)anc)[p];
  float w  = an.z - an.x, h  = an.w - an.y;
  float cx = an.x + 0.5f * w, cy = an.y + 0.5f * h;
  float px = dx * w + cx, py = dy * h + cy;
  float pw = expf(dw) * w, ph = expf(dh) * h;

  float4 o;
  o.x = fminf(fmaxf(px - 0.5f * pw, 0.f), IMGSZ);
  o.y = fminf(fmaxf(py - 0.5f * ph, 0.f), IMGSZ);
  o.z = fminf(fmaxf(px + 0.5f * pw, 0.f), IMGSZ);
  o.w = fminf(fmaxf(py + 0.5f * ph, 0.f), IMGSZ);
  boxes[(size_t)b * NPB + p] = o;
}

// ---------------------------------------------------------------------------
// Kernel 2: per-batch top-KPRE. Radix-select the threshold over 36864 scores,
// compact (key,~idx) pairs into LDS, bitonic-sort 2048 descending, gather
// the sorted top-2000 boxes contiguously into topBoxes.
// ---------------------------------------------------------------------------
__global__ __launch_bounds__(1024) void rpn_topk(const float* __restrict__ scores,
                                                 const float4* __restrict__ boxes,
                                                 float4* __restrict__ topBoxes) {
  __shared__ unsigned s_hist[256];
  __shared__ unsigned s_pref, s_krem, s_cA, s_cE;
  __shared__ unsigned long long s_sel[2048];

  int b = blockIdx.x, tid = threadIdx.x;
  const float* sc = scores + (size_t)b * NPB;

  if (tid == 0) { s_pref = 0u; s_krem = KPRE; }

  // 4x 8-bit radix passes, MSB first: find the KPRE-th largest sortable key.
  for (int shift = 24; shift >= 0; shift -= 8) {
    if (tid < 256) s_hist[tid] = 0u;
    __syncthreads();
    unsigned pref = s_pref;
    for (int p = tid; p < NPB; p += 1024) {
      unsigned key = f2key(sc[p]);
      bool ok = (shift == 24) || ((key >> (shift + 8)) == (pref >> (shift + 8)));
      if (ok) atomicAdd(&s_hist[(key >> shift) & 255u], 1u);
    }
    __syncthreads();
    if (tid == 0) {
      unsigned cum = 0, kr = s_krem; int dsel = 0;
      for (int d = 255; d >= 0; --d) {
        unsigned hv = s_hist[d];
        if (cum + hv >= kr) { dsel = d; break; }
        cum += hv;
      }
      s_pref = pref | ((unsigned)dsel << shift);
      s_krem = kr - cum;   // rank within the equal-to-threshold group
    }
    __syncthreads();
  }
  unsigned T      = s_pref;
  unsigned numEq  = s_krem;
  unsigned nAbove = KPRE - numEq;

  s_sel[tid] = 0ull; s_sel[tid + 1024] = 0ull;
  if (tid == 0) { s_cA = 0u; s_cE = 0u; }
  __syncthreads();

  // compact: strictly-above keys (exactly nAbove of them) + numEq equal keys
  for (int p = tid; p < NPB; p += 1024) {
    unsigned key = f2key(sc[p]);
    if (key > T) {
      unsigned pos = atomicAdd(&s_cA, 1u);
      s_sel[pos] = ((unsigned long long)key << 32) | (unsigned)(~(unsigned)p);
    } else if (key == T) {
      unsigned pos = atomicAdd(&s_cE, 1u);
      if (pos < numEq)
        s_sel[nAbove + pos] = ((unsigned long long)key << 32) | (unsigned)(~(unsigned)p);
    }
  }

  // bitonic sort 2048, descending on (key, ~idx): desc key, asc idx on ties.
  for (unsigned k = 2; k <= 2048; k <<= 1)
    for (unsigned j = k >> 1; j > 0; j >>= 1) {
      __syncthreads();
      unsigned t  = (unsigned)tid;
      unsigned i  = ((t & ~(j - 1)) << 1) | (t & (j - 1));
      unsigned pp = i | j;
      unsigned long long va = s_sel[i], vb = s_sel[pp];
      bool up = (i & k) == 0;
      bool sw = up ? (va < vb) : (va > vb);
      if (sw) { s_sel[i] = vb; s_sel[pp] = va; }
    }
  __syncthreads();

  for (int r = tid; r < KPRE; r += 1024) {
    unsigned idx = ~((unsigned)s_sel[r]);
    topBoxes[(size_t)b * KPRE + r] = boxes[(size_t)b * NPB + idx];
  }
}

// ---------------------------------------------------------------------------
// Kernel 3: per-batch greedy NMS over sorted top-2000, chunked 16-wide.
// Intra-chunk: wave 0 resolves the 16-box sequential chain via __shfl.
// Cross-chunk: one 16x16 tile per wave; union term area_i + area_j computed
// as a rank-2 outer product on the matrix pipe: V_WMMA_F32_16X16X4_F32 with
// A = [areas_i | 1 | 0 | 0] (16x4), B = [1 ; areas_j ; 0 ; 0] (4x16).
// Per-chunk row state (boxes + keep bits) is hoisted into registers /
// a ballot mask once per ci; the IoU test is a branchless mul-compare
// (uni >= 1e-9 > 0, so inter/uni > T  <=>  inter > T*uni).
// Boxes staged to LDS via async global->LDS loads when available.
// ---------------------------------------------------------------------------
__global__ __launch_bounds__(1024) void rpn_nms(const float4* __restrict__ topBoxes,
                                                unsigned* __restrict__ keepOut) {
  __shared__ float4   sBox[KPRE];    // 32 KB
  __shared__ float    sArea[KPRE];   //  8 KB
  __shared__ unsigned sKeep[KPRE];   //  8 KB

  int b = blockIdx.x, tid = threadIdx.x;
  int lane = tid & 31, wid = tid >> 5;
  const float4* tb = topBoxes + (size_t)b * KPRE;

  for (int t = tid; t < KPRE; t += 1024) {
#if USE_ASYNC_LDS
    __builtin_amdgcn_global_load_async_to_lds_b128(
        (gas_v4i_ptr)(tb + t),
        (las_v4i_ptr)(&sBox[t]), 0, 0);
#else
    sBox[t] = tb[t];
#endif
  }
#if USE_ASYNC_LDS
  __builtin_amdgcn_s_wait_asynccnt(0);
#endif
  __syncthreads();

  for (int t = tid; t < KPRE; t += 1024) {
    float4 bx = sBox[t];
    sArea[t] = fmaxf(bx.z - bx.x, 0.f) * fmaxf(bx.w - bx.y, 0.f);
    sKeep[t] = 1u;
  }
  __syncthreads();

  int hi = lane >> 4, lo = lane & 15;
  float hsel = (hi == 0) ? 1.f : 0.f;   // zero out K=2,3 half of the fragments

  for (int ci = 0; ci < NT_; ++ci) {
    // --- intra-chunk sequential resolution (wave 0, lockstep, no barrier) ---
    if (wid == 0) {
      int j  = ci * 16 + lo;
      float4 bj = sBox[j];
      float  aj = sArea[j];
      unsigned kf = sKeep[j];
      for (int ii = 0; ii < 15; ++ii) {
        unsigned ki = __shfl(kf, ii, 32);
        int i = ci * 16 + ii;
        float4 bi = sBox[i];
        float iw = fminf(bi.z, bj.z) - fmaxf(bi.x, bj.x);
        float ih = fminf(bi.w, bj.w) - fmaxf(bi.y, bj.y);
        float inter = fmaxf(iw, 0.f) * fmaxf(ih, 0.f);
        float uni = fmaxf(sArea[i] + aj - inter, 1e-9f);
        if (lo > ii && ki && (inter > IOUT * uni)) kf = 0u;
      }
      if (lane < 16) sKeep[j] = kf;
    }
    __syncthreads();

    // --- per-ci invariants: 16-bit keep mask + A fragment + row boxes ------
    unsigned kj = sKeep[ci * 16 + lo];                       // lanes 16-31 dup 0-15
    unsigned kmask = (unsigned)__ballot(kj != 0u) & 0xFFFFu; // wave-uniform
    v2f av;
    av.x = sArea[ci * 16 + lo] * hsel;   // A[m][0] = area_i[m] (K=0,1 half only)
    av.y = hsel;                         // A[m][1] = 1
    float4 bi[8];
#pragma unroll
    for (int r = 0; r < 8; ++r) bi[r] = sBox[ci * 16 + r + 8 * hi];

    // --- cross-chunk suppression: tiles (ci, cj>ci), one per wave ---------
    if (kmask) {
      for (int cj = ci + 1 + wid; cj < NT_; cj += 32) {
        int j = cj * 16 + lo;
        float4 bj = sBox[j];
        v2f bv;
        bv.x = hsel;                     // B[0][n] = 1
        bv.y = sArea[j] * hsel;          // B[1][n] = area_j[n]
        v8f acc = {0.f, 0.f, 0.f, 0.f, 0.f, 0.f, 0.f, 0.f};
        acc = __builtin_amdgcn_wmma_f32_16x16x4_f32(false, av, false, bv,
                                                    (short)0, acc, false, false);
        // D[m][n] = area_i[m] + area_j[n]; m = r + 8*hi, n = lo
        bool sup = false;
#pragma unroll
        for (int r = 0; r < 8; ++r) {
          int m = r + 8 * hi;
          float iw = fminf(bi[r].z, bj.z) - fmaxf(bi[r].x, bj.x);
          float ih = fminf(bi[r].w, bj.w) - fmaxf(bi[r].y, bj.y);
          float inter = fmaxf(iw, 0.f) * fmaxf(ih, 0.f);
          float uni = fmaxf(acc[r] - inter, 1e-9f);
          sup = sup || (((kmask >> m) & 1u) && (inter > IOUT * uni));
        }
        if (sup) sKeep[j] = 0u;          // at most one ds_store per tile per lane
      }
    }
    __syncthreads();
  }

  for (int t = tid; t < KPRE; t += 1024)
    keepOut[(size_t)b * KPRE + t] = sKeep[t];
}

// ---------------------------------------------------------------------------
// Kernel 4: final selection. kept&valid boxes (already score-descending)
// first, then the -inf slots in index order (JAX top_k tie rule); emit 1000.
// ---------------------------------------------------------------------------
__global__ __launch_bounds__(1024) void rpn_emit(const float4* __restrict__ topBoxes,
                                                 const unsigned* __restrict__ keep,
                                                 float* __restrict__ out) {
  __shared__ unsigned s_a[2048];
  int b = blockIdx.x, tid = threadIdx.x;
  const float4*   tb = topBoxes + (size_t)b * KPRE;
  const unsigned* kp = keep + (size_t)b * KPRE;

  int p0 = tid, p1 = tid + 1024;
  unsigned f0 = 0u, f1 = 0u;
  if (p0 < KPRE) {
    float4 bx = tb[p0];
    f0 = (kp[p0] && (bx.z - bx.x >= MINSZ) && (bx.w - bx.y >= MINSZ)) ? 1u : 0u;
  }
  if (p1 < KPRE) {
    float4 bx = tb[p1];
    f1 = (kp[p1] && (bx.z - bx.x >= MINSZ) && (bx.w - bx.y >= MINSZ)) ? 1u : 0u;
  }
  s_a[p0] = f0; s_a[p1] = f1;
  __syncthreads();

  // inclusive Hillis-Steele scan over 2048
  for (int off = 1; off < 2048; off <<= 1) {
    unsigned a0 = s_a[p0], a1 = s_a[p1];
    unsigned b0 = (p0 >= off) ? s_a[p0 - off] : 0u;
    unsigned b1 = (p1 >= off) ? s_a[p1 - off] : 0u;
    __syncthreads();
    s_a[p0] = a0 + b0; s_a[p1] = a1 + b1;
    __syncthreads();
  }
  unsigned C = s_a[KPRE - 1];   // total kept&valid

#pragma unroll
  for (int q = 0; q < 2; ++q) {
    int p = (q == 0) ? p0 : p1;
    unsigned f = (q == 0) ? f0 : f1;
    if (p < KPRE) {
      unsigned incl = s_a[p];
      unsigned e    = incl - f;             // exclusive rank among flagged
      unsigned pos  = f ? e : (C + (unsigned)p - e);
      if (pos < KPOST) {
        float4 bx = tb[p];
        float* o = out + ((size_t)b * KPOST + pos) * 4;
        o[0] = bx.x; o[1] = bx.y; o[2] = bx.z; o[3] = bx.w;
      }
    }
  }
}

// ---------------------------------------------------------------------------
extern "C" void kernel_launch(void* const* d_in, const int* in_sizes, int n_in,
                              void* d_out, int out_size, void* d_ws, size_t ws_size,
                              hipStream_t stream) {
  (void)in_sizes; (void)n_in; (void)out_size; (void)ws_size;
  const float* obj = (const float*)d_in[0];   // (8, 9, 64, 64)
  const float* del = (const float*)d_in[1];   // (8, 36, 64, 64)
  const float* anc = (const float*)d_in[2];   // (36864, 4)
  float* out = (float*)d_out;                 // (8, 1000, 4)

  // workspace layout (~6.2 MB)
  char* ws = (char*)d_ws;
  float4*   wsBoxes  = (float4*)ws;                                   // 8*36864*16 B
  float*    wsScores = (float*)(ws + (size_t)B_ * NPB * 16);          // 8*36864*4 B
  float4*   wsTop    = (float4*)(ws + (size_t)B_ * NPB * 16 +
                                 (size_t)B_ * NPB * 4);               // 8*2000*16 B
  unsigned* wsKeep   = (unsigned*)((char*)wsTop + (size_t)B_ * KPRE * 16);

  dim3 g1(B_, NPB / 256);
  rpn_decode<<<g1, 256, 0, stream>>>(obj, del, anc, wsBoxes, wsScores);
  rpn_topk<<<B_, 1024, 0, stream>>>(wsScores, wsBoxes, wsTop);
  rpn_nms<<<B_, 1024, 0, stream>>>(wsTop, wsKeep);
  rpn_emit<<<B_, 1024, 0, stream>>>(wsTop, wsKeep, out);
}